// SWF2LUT_58918361367050
// MI455X (gfx1250) — compile-verified
//
#include <hip/hip_runtime.h>

// ---------------------------------------------------------------------------
// Quadrilinear (4-simplex) LUT interpolation, SWF2LUT-style.
//   Q = 16, L = 17, LUT = (L^4, 3) float weights, quantized to int8.
//   img: (8,1,513,513) f32 holding exact ints 0..255
//   out: (8,1,512,512,3) f32
// Strategy: int8-quantized LUT (250,563 B) staged into LDS (320 KB/WGP on
// CDNA5) via async global->LDS b128 loads (ASYNCcnt + s_wait_asynccnt);
// per-pixel: 4 loads, 5-comparator sort network, 15 LDS byte gathers,
// exact integer MAC, *1/16.
// ---------------------------------------------------------------------------

#define QV 16
#define LV 17
#define L2V (LV * LV)                  // 289
#define L3V (LV * LV * LV)             // 4913
#define LUT_ENTRIES (L3V * LV)         // 83521
#define LUT_BYTES (LUT_ENTRIES * 3)    // 250563
#define LUT_BYTES_PAD ((LUT_BYTES + 15) & ~15)

#define IMG_N 8
#define OUT_H 512
#define OUT_W 512
#define NPIX (IMG_N * OUT_H * OUT_W)   // 2,097,152
#define IN_W 513
#define IN_IMG_STRIDE (IN_W * IN_W)    // 263169

#define BLOCK 1024
#define NBLOCKS 256

typedef int v4i __attribute__((ext_vector_type(4)));
typedef __attribute__((address_space(1))) v4i*  gptr_v4i;
typedef __attribute__((address_space(3))) v4i*  lptr_v4i;
typedef __attribute__((address_space(1))) char* gptr_b;
typedef __attribute__((address_space(3))) char* lptr_b;

// ---- kernel 1: quantize float weights -> int8 LUT in workspace -------------
__global__ void SWF2LUT_quantize(const float* __restrict__ w,
                                 signed char* __restrict__ w8, int n) {
    int i = blockIdx.x * blockDim.x + threadIdx.x;
    if (i < n) {
        float v = rintf(w[i] * 127.0f);          // round-half-even, like jnp.round
        v = fminf(127.0f, fmaxf(-127.0f, v));
        w8[i] = (signed char)v;
    }
}

// ---- kernel 2: main interpolation ------------------------------------------
__global__ __launch_bounds__(BLOCK) void SWF2LUT_interp(
        const float* __restrict__ img,
        const signed char* __restrict__ w8,
        float* __restrict__ out) {
    extern __shared__ signed char lut[];

    // ---- stage the int8 LUT into LDS ----
    const int nvec = LUT_BYTES / 16;             // 15660 x b128
#if __has_builtin(__builtin_amdgcn_global_load_async_to_lds_b128)
    {
        gptr_v4i gsrc = (gptr_v4i)w8;
        lptr_v4i ldst = (lptr_v4i)lut;
        for (int i = threadIdx.x; i < nvec; i += BLOCK) {
            __builtin_amdgcn_global_load_async_to_lds_b128(gsrc + i, ldst + i, 0, 0);
        }
        // tail (250563 % 16 == 3 bytes)
#if __has_builtin(__builtin_amdgcn_global_load_async_to_lds_b8)
        for (int i = nvec * 16 + (int)threadIdx.x; i < LUT_BYTES; i += BLOCK) {
            __builtin_amdgcn_global_load_async_to_lds_b8(
                (gptr_b)w8 + i, (lptr_b)lut + i, 0, 0);
        }
#else
        for (int i = nvec * 16 + (int)threadIdx.x; i < LUT_BYTES; i += BLOCK)
            lut[i] = w8[i];
#endif
#if __has_builtin(__builtin_amdgcn_s_wait_asynccnt)
        __builtin_amdgcn_s_wait_asynccnt(0);
#else
        asm volatile("s_wait_asynccnt 0" ::: "memory");
#endif
    }
#else
    {
        for (int i = threadIdx.x; i < nvec; i += BLOCK)
            ((int4*)lut)[i] = ((const int4*)w8)[i];
        for (int i = nvec * 16 + (int)threadIdx.x; i < LUT_BYTES; i += BLOCK)
            lut[i] = w8[i];
    }
#endif
    __syncthreads();

    // ---- grid-stride over output pixels ----
    const int tid = blockIdx.x * BLOCK + threadIdx.x;
    const int nthreads = gridDim.x * BLOCK;
    for (int p = tid; p < NPIX; p += nthreads) {
        const int n   = p / (OUT_H * OUT_W);
        const int rem = p - n * (OUT_H * OUT_W);
        const int y   = rem / OUT_W;
        const int x   = rem - y * OUT_W;

        const float* bp = img + (size_t)n * IN_IMG_STRIDE + (size_t)y * IN_W + x;
        const int ai = (int)bp[0];
        const int bi = (int)bp[1];
        const int ci = (int)bp[IN_W];
        const int di = (int)bp[IN_W + 1];

        const int ia = ai >> 4, ib = bi >> 4, ic = ci >> 4, id = di >> 4;

        // key = frac*4 + position  (exactly the reference's tie-break)
        int k0 = ((ai & 15) << 2) | 0;  int s0 = L3V;
        int k1 = ((bi & 15) << 2) | 1;  int s1 = L2V;
        int k2 = ((ci & 15) << 2) | 2;  int s2 = LV;
        int k3 = ((di & 15) << 2) | 3;  int s3 = 1;

        // descending sort of 4 distinct keys: network (0,1)(2,3)(0,2)(1,3)(1,2)
#define CSWAP(i, j)                                                      \
        if (k##i < k##j) { int tk = k##i; k##i = k##j; k##j = tk;        \
                           int ts = s##i; s##i = s##j; s##j = ts; }
        CSWAP(0, 1) CSWAP(2, 3) CSWAP(0, 2) CSWAP(1, 3) CSWAP(1, 2)
#undef CSWAP

        const int fs0 = k0 >> 2, fs1 = k1 >> 2, fs2 = k2 >> 2, fs3 = k3 >> 2;

        int i0 = ia * L3V + ib * L2V + ic * LV + id;
        int i1 = i0 + s0;
        int i2 = i1 + s1;
        int i3 = i2 + s2;
        int i4 = i3 + s3;
        i0 *= 3; i1 *= 3; i2 *= 3; i3 *= 3; i4 *= 3;

        const int w0 = QV - fs0;
        const int w1 = fs0 - fs1;
        const int w2 = fs1 - fs2;
        const int w3 = fs2 - fs3;
        const int w4 = fs3;

        float* op = out + (size_t)p * 3;
#pragma unroll
        for (int ch = 0; ch < 3; ++ch) {
            const int acc = w0 * (int)lut[i0 + ch] + w1 * (int)lut[i1 + ch] +
                            w2 * (int)lut[i2 + ch] + w3 * (int)lut[i3 + ch] +
                            w4 * (int)lut[i4 + ch];
            op[ch] = (float)acc * 0.0625f;   // exact: /Q
        }
    }
}

// ---------------------------------------------------------------------------
extern "C" void kernel_launch(void* const* d_in, const int* in_sizes, int n_in,
                              void* d_out, int out_size, void* d_ws, size_t ws_size,
                              hipStream_t stream) {
    const float* img    = (const float*)d_in[0];    // (8,1,513,513) f32
    const float* weight = (const float*)d_in[1];    // (83521, 3)    f32
    float* out          = (float*)d_out;            // (8,1,512,512,3) f32
    signed char* w8     = (signed char*)d_ws;       // int8 LUT scratch

    // allow >64KB dynamic LDS (idempotent, capture-safe, no static guards)
    (void)hipFuncSetAttribute((const void*)SWF2LUT_interp,
                              hipFuncAttributeMaxDynamicSharedMemorySize,
                              LUT_BYTES_PAD);

    const int nw = in_sizes[1];                     // 250563
    SWF2LUT_quantize<<<(nw + 255) / 256, 256, 0, stream>>>(weight, w8, nw);
    SWF2LUT_interp<<<NBLOCKS, BLOCK, LUT_BYTES_PAD, stream>>>(img, w8, out);
}